// SCRFDPostModel_16956530885001
// MI455X (gfx1250) — compile-verified
//
#include <hip/hip_runtime.h>
#include <cstdint>

// SCRFD post-process decode for MI455X (gfx1250).
//
// Memory-bound streaming kernel (~516 MB round trip, floor ~22us @ 23.3 TB/s).
// Uses CDNA5 async global<->LDS DMA (ASYNCcnt) to turn the stride-40B (kp) and
// stride-56B (output) per-anchor records into perfectly sequential b128 streams.
// All streaming traffic is marked non-temporal: the 516 MB read-once/write-once
// working set exceeds the 192 MB L2, so NT avoids thrashing the cache.

namespace {

constexpr int TILE        = 128;     // anchors per block, one per thread
constexpr int BATCH       = 64;
constexpr int NTOT        = 67200;   // 51200 + 12800 + 3200
constexpr int TILES_PER_B = 525;     // 400 + 100 + 25

__device__ __forceinline__ uint32_t lds_addr(const void* p) {
  // generic pointer to __shared__ : low 32 bits are the LDS byte offset
  return (uint32_t)(uintptr_t)p;
}

__device__ __forceinline__ void async_load_b128(uint32_t lds_off,
                                                const void* gbase,
                                                uint32_t voff) {
  // GVS mode: saddr = 64-bit base, vaddr = 32-bit byte offset, vdst = LDS addr
  asm volatile("global_load_async_to_lds_b128 %0, %1, %2 th:TH_LOAD_NT"
               :: "v"(lds_off), "v"(voff), "s"(gbase) : "memory");
}

__device__ __forceinline__ void async_store_b128(const void* gbase,
                                                 uint32_t voff,
                                                 uint32_t lds_off) {
  asm volatile("global_store_async_from_lds_b128 %0, %1, %2 th:TH_STORE_NT"
               :: "v"(voff), "v"(lds_off), "s"(gbase) : "memory");
}

__device__ __forceinline__ void wait_asynccnt0() {
  asm volatile("s_wait_asynccnt 0" ::: "memory");
}

__global__ __launch_bounds__(TILE)
void scrfd_post_kernel(const float* __restrict__ cls0, const float* __restrict__ bbox0, const float* __restrict__ kp0,
                       const float* __restrict__ cls1, const float* __restrict__ bbox1, const float* __restrict__ kp1,
                       const float* __restrict__ cls2, const float* __restrict__ bbox2, const float* __restrict__ kp2,
                       const float* __restrict__ shapes, float* __restrict__ out)
{
  __shared__ __align__(16) float s_cls[TILE];        //  512 B
  __shared__ __align__(16) float s_bbox[TILE * 4];   // 2048 B
  __shared__ __align__(16) float s_kp[TILE * 10];    // 5120 B
  __shared__ __align__(16) float s_out[TILE * 14];   // 7168 B

  const int tid  = threadIdx.x;
  const int b    = blockIdx.x / TILES_PER_B;
  const int tile = blockIdx.x % TILES_PER_B;

  // Every level size (51200/12800/3200) is a multiple of TILE=128,
  // so a tile never straddles a level boundary.
  const float *cls, *bbox, *kp;
  int n0, strd, W, Nl;
  if (tile < 400)      { cls = cls0; bbox = bbox0; kp = kp0; n0 = tile * TILE;         strd = 8;  W = 160; Nl = 51200; }
  else if (tile < 500) { cls = cls1; bbox = bbox1; kp = kp1; n0 = (tile - 400) * TILE; strd = 16; W = 80;  Nl = 12800; }
  else                 { cls = cls2; bbox = bbox2; kp = kp2; n0 = (tile - 500) * TILE; strd = 32; W = 40;  Nl = 3200;  }

  const size_t base = (size_t)b * (size_t)Nl + (size_t)n0;
  const char* gcls  = (const char*)(cls  + base);        // 16B aligned (base % 4  == 0)
  const char* gbbox = (const char*)(bbox + base * 4);    // 16B aligned always
  const char* gkp   = (const char*)(kp   + base * 10);   // 16B aligned (base even)

  const uint32_t l_cls  = lds_addr(s_cls);
  const uint32_t l_bbox = lds_addr(s_bbox);
  const uint32_t l_kp   = lds_addr(s_kp);
  const uint32_t l_out  = lds_addr(s_out);

  // ---- async DMA tile in: 512 + 2048 + 5120 bytes as b128 granules ----
  if (tid < 32)
    async_load_b128(l_cls + (uint32_t)tid * 16u, gcls, (uint32_t)tid * 16u);
  async_load_b128(l_bbox + (uint32_t)tid * 16u, gbbox, (uint32_t)tid * 16u);
  async_load_b128(l_kp + (uint32_t)tid * 16u,                gkp, (uint32_t)tid * 16u);
  async_load_b128(l_kp + (uint32_t)(TILE + tid) * 16u,       gkp, (uint32_t)(TILE + tid) * 16u);
  if (tid < 64)
    async_load_b128(l_kp + (uint32_t)(2 * TILE + tid) * 16u, gkp, (uint32_t)(2 * TILE + tid) * 16u);
  wait_asynccnt0();
  __syncthreads();

  // ---- per-anchor decode in registers ----
  const int   n  = n0 + tid;
  const int   p  = n >> 1;               // A = 2 anchors per location
  const float fs = (float)strd;
  const float cx = (float)((p % W) * strd);
  const float cy = (float)((p / W) * strd);
  // ratio_wh = (origin / 1280)[::-1] : x scaled by shapes[:,1], y by shapes[:,0]
  const float rx = shapes[2 * b + 1] * (1.0f / 1280.0f);
  const float ry = shapes[2 * b + 0] * (1.0f / 1280.0f);

  const float sc = 1.0f / (1.0f + __expf(-s_cls[tid]));
  const bool keep = sc > 0.05f;

  const float4 bb = *reinterpret_cast<const float4*>(&s_bbox[tid * 4]);
  const float x1 = cx - bb.x * fs;
  const float y1 = cy - bb.y * fs;
  const float x2 = cx + bb.z * fs;
  const float y2 = cy + bb.w * fs;

  float* o = &s_out[tid * 14];
  o[0] = keep ? x1 * rx : -1.0f;
  o[1] = keep ? y1 * ry : -1.0f;
  o[2] = keep ? x2 * rx : -1.0f;
  o[3] = keep ? y2 * ry : -1.0f;
#pragma unroll
  for (int k = 0; k < 5; ++k) {
    const float kx = cx + s_kp[tid * 10 + 2 * k    ] * fs;
    const float ky = cy + s_kp[tid * 10 + 2 * k + 1] * fs;
    o[4 + 2 * k] = keep ? kx * rx : -1.0f;
    o[5 + 2 * k] = keep ? ky * ry : -1.0f;
  }
  __syncthreads();   // (includes dscnt wait) all LDS output written

  // ---- async DMA tile out: 7168 B = 448 b128 granules ----
  // out anchor index in concatenated space is tile*TILE (400*128 == 51200 etc.)
  // base byte offset 56 * (b*NTOT + tile*128) is a multiple of 16.
  const char* gout = (const char*)(out + ((size_t)b * NTOT + (size_t)tile * TILE) * 14);
  async_store_b128(gout, (uint32_t)tid * 16u,              l_out + (uint32_t)tid * 16u);
  async_store_b128(gout, (uint32_t)(TILE + tid) * 16u,     l_out + (uint32_t)(TILE + tid) * 16u);
  async_store_b128(gout, (uint32_t)(2 * TILE + tid) * 16u, l_out + (uint32_t)(2 * TILE + tid) * 16u);
  if (tid < 64)
    async_store_b128(gout, (uint32_t)(3 * TILE + tid) * 16u, l_out + (uint32_t)(3 * TILE + tid) * 16u);

  // scores: second output tensor, raw (unmasked), coalesced NT b32 stream
  __builtin_nontemporal_store(
      sc, &out[(size_t)BATCH * NTOT * 14 + (size_t)b * NTOT + (size_t)tile * TILE + tid]);

  wait_asynccnt0();  // (S_ENDPGM also implies wait-idle)
}

} // namespace

extern "C" void kernel_launch(void* const* d_in, const int* in_sizes, int n_in,
                              void* d_out, int out_size, void* d_ws, size_t ws_size,
                              hipStream_t stream) {
  (void)in_sizes; (void)n_in; (void)out_size; (void)d_ws; (void)ws_size;
  dim3 grid(BATCH * TILES_PER_B);   // 33,600 blocks
  dim3 block(TILE);                 // 128 threads = 4 wave32
  scrfd_post_kernel<<<grid, block, 0, stream>>>(
      (const float*)d_in[0], (const float*)d_in[1], (const float*)d_in[2],
      (const float*)d_in[3], (const float*)d_in[4], (const float*)d_in[5],
      (const float*)d_in[6], (const float*)d_in[7], (const float*)d_in[8],
      (const float*)d_in[9], (float*)d_out);
}